// MeshTransformer_42253888258487
// MI455X (gfx1250) — compile-verified
//
#include <hip/hip_runtime.h>

typedef __attribute__((ext_vector_type(2))) float v2f;
typedef __attribute__((ext_vector_type(8))) float v8f;

// Problem constants (B=16, S=16, P=64, V=2562)
constexpr int kBS    = 256;               // B*S
constexpr int kP     = 64;                // prototypes
constexpr int kV     = 2562;              // vertices
constexpr int kK     = kP * 3;            // 192  GEMM K
constexpr int kKP    = kK / 2;            // 96   packed K row-pairs
constexpr int kM     = kBS * 3;           // 768  GEMM M
constexpr int kMT    = kM / 16;           // 48   M tiles
constexpr int kNTILE = 4;                 // N tiles per wave
constexpr int kNT4   = 41;                // ceil(2562 / 64)
constexpr int kNPAD  = kNT4 * kNTILE * 16;// 2624 padded N (zero-filled tail)

// --- Kernel 1: A[(bs*3+i)*K + p*3+j] = (Rx@Ry@Rz)[i][j] * w[bs,p] * scale[bs]
__global__ void prep_A_kernel(const float* __restrict__ scales,
                              const float* __restrict__ transforms,
                              const float* __restrict__ pw,
                              float* __restrict__ A) {
  int idx = blockIdx.x * blockDim.x + threadIdx.x;  // over BS*P
  if (idx >= kBS * kP) return;
  int bs = idx >> 6;
  int p  = idx & 63;
  const float* t = transforms + (size_t)idx * 6;
  float ax = t[3], ay = t[4], az = t[5];
  float cx = cosf(ax), sx = sinf(ax);
  float cy = cosf(ay), sy = sinf(ay);
  float cz = cosf(az), sz = sinf(az);
  // R = Rx(ax) @ Ry(ay) @ Rz(az)
  float r00 = cy * cz;
  float r01 = -cy * sz;
  float r02 = sy;
  float r10 = sx * sy * cz + cx * sz;
  float r11 = -sx * sy * sz + cx * cz;
  float r12 = -sx * cy;
  float r20 = -cx * sy * cz + sx * sz;
  float r21 = cx * sy * sz + sx * cz;
  float r22 = cx * cy;
  float w = pw[idx] * scales[bs];
  float* a0 = A + (size_t)(bs * 3 + 0) * kK + p * 3;
  float* a1 = A + (size_t)(bs * 3 + 1) * kK + p * 3;
  float* a2 = A + (size_t)(bs * 3 + 2) * kK + p * 3;
  a0[0] = r00 * w; a0[1] = r01 * w; a0[2] = r02 * w;
  a1[0] = r10 * w; a1[1] = r11 * w; a1[2] = r12 * w;
  a2[0] = r20 * w; a2[1] = r21 * w; a2[2] = r22 * w;
}

// --- Kernel 2: packed row-pair B: Dp[kp*NPAD + n] = {D[2kp][n], D[2kp+1][n]}
//     where D[k][n] = base[n,j] + off[p,n,j], k=(p,j); zero pad n>=V.
__global__ void prep_D_kernel(const float* __restrict__ base,
                              const float* __restrict__ off,
                              v2f* __restrict__ Dp) {
  int id = blockIdx.x * blockDim.x + threadIdx.x;
  if (id >= kKP * kNPAD) return;
  int kp = id / kNPAD;
  int n  = id - kp * kNPAD;
  v2f val = {0.f, 0.f};
  if (n < kV) {
    int k0 = kp * 2, k1 = k0 + 1;
    int p0 = k0 / 3, j0 = k0 - p0 * 3;
    int p1 = k1 / 3, j1 = k1 - p1 * 3;
    val.x = base[(size_t)n * 3 + j0] + off[((size_t)p0 * kV + n) * 3 + j0];
    val.y = base[(size_t)n * 3 + j1] + off[((size_t)p1 * kV + n) * 3 + j1];
  }
  Dp[id] = val;
}

// --- Kernel 3: T[bs*3+i] = sum_p w[bs,p] * translation[bs,p,i]
__global__ void prep_T_kernel(const float* __restrict__ transforms,
                              const float* __restrict__ pw,
                              float* __restrict__ T) {
  __shared__ float sm[kP][3];
  int bs = blockIdx.x;
  int p  = threadIdx.x;  // 0..63
  const float* t = transforms + ((size_t)bs * kP + p) * 6;
  float w = pw[bs * kP + p];
  sm[p][0] = w * t[0];
  sm[p][1] = w * t[1];
  sm[p][2] = w * t[2];
  __syncthreads();
  for (int st = 32; st > 0; st >>= 1) {
    if (p < st) {
      sm[p][0] += sm[p + st][0];
      sm[p][1] += sm[p + st][1];
      sm[p][2] += sm[p + st][2];
    }
    __syncthreads();
  }
  if (p < 3) T[bs * 3 + p] = sm[0][p];
}

// --- Kernel 4: O[768 x 2562] = A[768 x 192] * D[192 x 2624] via f32 WMMA.
//     One wave per (16M x 64N): 4 accumulators share each A fragment.
//     Epilogue adds T[m] and scatters into the (bs, v, i) output layout.
__global__ __launch_bounds__(32) void gemm_wmma_kernel(
    const float* __restrict__ A, const v2f* __restrict__ Dp,
    const float* __restrict__ T, float* __restrict__ out) {
  int lane = threadIdx.x;       // 0..31, one wave per block
  int mt = blockIdx.x;          // 0..47
  int nt = blockIdx.y;          // 0..40
  int m0 = mt * 16;
  int n0 = nt * (kNTILE * 16);
  int half = lane >> 4;         // selects K pair {k,k+1} vs {k+2,k+3}
  int lrow = lane & 15;

  // A frag: lane holds A[m0+lrow][k + half*2 + {0,1}] (one b64 load)
  const float* Ap = A + (size_t)(m0 + lrow) * kK + half * 2;
  // B frag: lane holds row-pair (k/2 + half) at column n (one b64 load)
  const v2f* Bp = Dp + (size_t)half * kNPAD + n0 + lrow;

  v8f c0 = {0.f, 0.f, 0.f, 0.f, 0.f, 0.f, 0.f, 0.f};
  v8f c1 = c0, c2 = c0, c3 = c0;
#pragma unroll 4
  for (int k = 0; k < kK; k += 4) {
    v2f a = *(const v2f*)(Ap + k);
    const v2f* b = Bp + (size_t)(k >> 1) * kNPAD;
    v2f b0 = b[0], b1 = b[16], b2 = b[32], b3 = b[48];
    // (neg_a, A, neg_b, B, c_mod, C, reuse_a, reuse_b)
    c0 = __builtin_amdgcn_wmma_f32_16x16x4_f32(false, a, false, b0, (short)0,
                                               c0, false, false);
    c1 = __builtin_amdgcn_wmma_f32_16x16x4_f32(false, a, false, b1, (short)0,
                                               c1, false, false);
    c2 = __builtin_amdgcn_wmma_f32_16x16x4_f32(false, a, false, b2, (short)0,
                                               c2, false, false);
    c3 = __builtin_amdgcn_wmma_f32_16x16x4_f32(false, a, false, b3, (short)0,
                                               c3, false, false);
  }

  // Epilogue: C/D layout -> VGPR r, lanes 0-15: M=m0+r, lanes 16-31: M=m0+r+8.
  // m = bs*3 + i; advance (bs, i) incrementally (single division).
  int mb = m0 + half * 8;
  int bs = mb / 3;
  int i  = mb - bs * 3;
  float  tv[8];
  size_t ob[8];  // out base: bs*V*3 + i  (add n*3 per column)
#pragma unroll
  for (int r = 0; r < 8; ++r) {
    tv[r] = T[mb + r];
    ob[r] = (size_t)bs * (kV * 3) + i;
    if (++i == 3) { i = 0; ++bs; }
  }
  auto store_tile = [&](int t, const v8f& c) {
    int n = n0 + t * 16 + lrow;  // vertex index
    if (n < kV) {
      size_t n3 = (size_t)n * 3;
#pragma unroll
      for (int r = 0; r < 8; ++r) out[ob[r] + n3] = c[r] + tv[r];
    }
  };
  store_tile(0, c0);
  store_tile(1, c1);
  store_tile(2, c2);
  store_tile(3, c3);
}

extern "C" void kernel_launch(void* const* d_in, const int* in_sizes, int n_in,
                              void* d_out, int out_size, void* d_ws,
                              size_t ws_size, hipStream_t stream) {
  const float* scales     = (const float*)d_in[0];  // (B,S,1)
  const float* transforms = (const float*)d_in[1];  // (B,S,P,6)
  const float* pw         = (const float*)d_in[2];  // (B,S,P)
  const float* off        = (const float*)d_in[3];  // (P,V,3)
  const float* base       = (const float*)d_in[4];  // (V,3)
  float* out = (float*)d_out;

  float* wsf = (float*)d_ws;
  float* A  = wsf;                              // 768*192  = 147456 floats
  v2f*   Dp = (v2f*)(wsf + (size_t)kM * kK);    // 96*2624  = 251904 v2f (2 MB)
  float* T  = (float*)(Dp + (size_t)kKP * kNPAD);  // 768 floats

  prep_A_kernel<<<(kBS * kP + 255) / 256, 256, 0, stream>>>(scales, transforms,
                                                            pw, A);
  prep_D_kernel<<<(kKP * kNPAD + 255) / 256, 256, 0, stream>>>(base, off, Dp);
  prep_T_kernel<<<kBS, kP, 0, stream>>>(transforms, pw, T);
  gemm_wmma_kernel<<<dim3(kMT, kNT4), 32, 0, stream>>>(A, Dp, T, out);
}